// I0_32323923870519
// MI455X (gfx1250) — compile-verified
//
#include <hip/hip_runtime.h>

// Elementwise modified Bessel I0 on fp32.
// Memory-bound: 2^28 elems * 8 B = 2.15 GB moved -> ~92 us floor at 23.3 TB/s.
// Input is uniform [0,1): on that range the Maclaurin series
//   i0(x) = sum_k (x^2/4)^k / (k!)^2,  k=0..6
// has truncation error ~2.4e-12 (<< fp32 ulp), so the hot path is a degree-6
// Horner in t = x^2: 1 mul + 6 FMA = 7 VALU ops/element, ~4x cheaper than the
// Cephes Chebyshev evaluation and safely under the HBM roofline.
// A full Cephes/JAX-compatible path is kept as a cold, execz-skipped fixup for
// |x| >= 1 so the kernel stays correct for arbitrary inputs.

typedef float v4f __attribute__((ext_vector_type(4)));

// ---- hot path: |x| < 1, argument is t = x*x -------------------------------
__device__ __forceinline__ float i0_taylor(float t) {
    // c_k = 1 / (4^k * (k!)^2)
    float p =         4.7094178826e-10f;   // k=6
    p = fmaf(p, t,    6.7816840278e-8f);   // k=5
    p = fmaf(p, t,    6.7816840278e-6f);   // k=4
    p = fmaf(p, t,    4.3402777778e-4f);   // k=3
    p = fmaf(p, t,    1.5625e-2f);         // k=2
    p = fmaf(p, t,    0.25f);              // k=1
    p = fmaf(p, t,    1.0f);               // k=0
    return p;
}

// ---- cold path: Cephes single-precision i0 (|x| >= 1) ---------------------
__device__ __noinline__ float i0_cephes(float x) {
    x = __builtin_fabsf(x);
    if (x <= 8.0f) {
        const float A[18] = {
            -1.30002500998624804212e-8f,  6.04699502254191894932e-8f,
            -2.67079385394061173391e-7f,  1.11738753912010371815e-6f,
            -4.41673835845875056359e-6f,  1.64484480707288970893e-5f,
            -5.75419501008210370398e-5f,  1.88502885095841655729e-4f,
            -5.76375574538582365885e-4f,  1.63947561694133579842e-3f,
            -4.32430999505057594430e-3f,  1.05464603945949983183e-2f,
            -2.37374148058994688156e-2f,  4.93052842396707084878e-2f,
            -9.49010970480476444210e-2f,  1.71620901522208775349e-1f,
            -3.04682672343198398683e-1f,  6.76795274409476084995e-1f
        };
        const float y = fmaf(x, 0.5f, -2.0f);
        float b0 = A[0], b1 = 0.0f, b2 = 0.0f;
#pragma unroll
        for (int i = 1; i < 18; ++i) {
            b2 = b1; b1 = b0;
            b0 = fmaf(y, b1, A[i] - b2);
        }
        return 0.5f * (b0 - b2) * __expf(x);
    } else {
        const float B[7] = {
            3.39623202570838634515e-9f, 2.26666899049817806459e-8f,
            2.04891858946906374183e-7f, 2.89137052083475648297e-6f,
            6.88975834691682398426e-5f, 3.36911647825569408990e-3f,
            8.04490411014108831608e-1f
        };
        const float y = fmaf(32.0f, __frcp_rn(x), -2.0f);
        float b0 = B[0], b1 = 0.0f, b2 = 0.0f;
#pragma unroll
        for (int i = 1; i < 7; ++i) {
            b2 = b1; b1 = b0;
            b0 = fmaf(y, b1, B[i] - b2);
        }
        return 0.5f * (b0 - b2) * __expf(x) * __frsqrt_rn(x);
    }
}

__device__ __forceinline__ float i0_scalar(float x) {
    const float t = x * x;
    if (t < 1.0f) return i0_taylor(t);
    return i0_cephes(x);
}

__global__ __launch_bounds__(256) void i0_elementwise_kernel(
    const float* __restrict__ in, float* __restrict__ out, long long n) {
    const long long i4 = ((long long)blockIdx.x * blockDim.x + threadIdx.x) * 4;
    if (i4 + 3 < n) {
        // global_load_b128 th:NT  (2 GiB stream >> 192 MB L2: don't cache)
        v4f v = __builtin_nontemporal_load(reinterpret_cast<const v4f*>(in + i4));

        // t = x^2 (sign-independent: also serves as the |x|<1 range check)
        const v4f t = v * v;

        // Four independent 6-FMA Horner chains -> good dual-issue material.
        v4f r;
        r.x = i0_taylor(t.x);
        r.y = i0_taylor(t.y);
        r.z = i0_taylor(t.z);
        r.w = i0_taylor(t.w);

        // Single wave-level guard for the cold Cephes fixup; never taken for
        // uniform [0,1) input -> s_cbranch_execz skips the whole block.
        const float tmax = __builtin_fmaxf(__builtin_fmaxf(t.x, t.y),
                                           __builtin_fmaxf(t.z, t.w));
        if (__builtin_expect(!(tmax < 1.0f), 0)) {
            if (!(t.x < 1.0f)) r.x = i0_cephes(v.x);
            if (!(t.y < 1.0f)) r.y = i0_cephes(v.y);
            if (!(t.z < 1.0f)) r.z = i0_cephes(v.z);
            if (!(t.w < 1.0f)) r.w = i0_cephes(v.w);
        }

        // global_store_b128 th:NT
        __builtin_nontemporal_store(r, reinterpret_cast<v4f*>(out + i4));
    } else {
        // Scalar tail (n = 2^28 is divisible by 4, normally dead)
        for (long long i = i4; i < n; ++i) {
            float x = __builtin_nontemporal_load(in + i);
            __builtin_nontemporal_store(i0_scalar(x), out + i);
        }
    }
}

extern "C" void kernel_launch(void* const* d_in, const int* in_sizes, int n_in,
                              void* d_out, int out_size, void* d_ws, size_t ws_size,
                              hipStream_t stream) {
    (void)n_in; (void)d_ws; (void)ws_size;
    const float* x = (const float*)d_in[0];
    float* out = (float*)d_out;
    const long long n = (long long)out_size;         // 268,435,456 elements
    const long long n4 = (n + 3) / 4;                // float4 work-items
    const int block = 256;                           // 8 wave32s per block
    const long long grid = (n4 + block - 1) / block; // 262,144 blocks
    i0_elementwise_kernel<<<(unsigned)grid, block, 0, stream>>>(x, out, n);
}